// WeldonModel_29755533427374
// MI455X (gfx1250) — compile-verified
//
#include <hip/hip_runtime.h>
#include <hip/hip_bf16.h>

typedef __attribute__((ext_vector_type(2))) float v2f;
typedef __attribute__((ext_vector_type(8))) float v8f;

#define BB 32
#define LL 4000
#define DD 2048
#define RR 10
#define WAVES_PER_BLOCK 8   // 256 threads = 8 wave32

// ---------------------------------------------------------------------------
// Stage 1: scores[b,l] = dot(x[b,l,:], w)  via V_WMMA_F32_16X16X4_F32.
// One wave handles a 16-row tile. A = 16 rows of x (16x4 f32 fragment:
// lanes 0-15 carry K=0,1 ; lanes 16-31 carry K=2,3). B = w[k..k+3]
// broadcast across all 16 columns (same per-lane k-offset pattern).
// After accumulating over K=2048, every column of D equals the row's dot
// product; lane 0 holds rows 0..7 (VGPR 0..7), lane 16 holds rows 8..15.
// ---------------------------------------------------------------------------
__global__ __launch_bounds__(256) void weldon_gemv_wmma(
    const float* __restrict__ x, const float* __restrict__ w,
    float* __restrict__ scores)
{
    const int lane = threadIdx.x & 31;
    const int wave = threadIdx.x >> 5;
    const int tile = blockIdx.x * WAVES_PER_BLOCK + wave;  // 8000 tiles
    const long long row_base = (long long)tile * 16;       // flat row = b*LL + l
    const int lane_row = lane & 15;
    const int koff = (lane >> 4) << 1;                     // 0 for lanes<16, 2 else

    const float* px = x + (row_base + lane_row) * (long long)DD + koff;
    const float* pw = w + koff;

    v8f c0 = {};
    v8f c1 = {};

    for (int k = 0; k < DD; k += 8) {
        if ((k & 255) == 0) {
            // stay ahead of the 8KB-strided row streams (global_prefetch_b8)
            __builtin_prefetch(px + k + 512, 0, 1);
        }
        v2f a0 = *reinterpret_cast<const v2f*>(px + k);
        v2f b0 = *reinterpret_cast<const v2f*>(pw + k);
        c0 = __builtin_amdgcn_wmma_f32_16x16x4_f32(
                 /*neg_a=*/false, a0, /*neg_b=*/false, b0,
                 /*c_mod=*/(short)0, c0, /*reuse_a=*/false, /*reuse_b=*/false);

        v2f a1 = *reinterpret_cast<const v2f*>(px + k + 4);
        v2f b1 = *reinterpret_cast<const v2f*>(pw + k + 4);
        c1 = __builtin_amdgcn_wmma_f32_16x16x4_f32(
                 false, a1, false, b1, (short)0, c1, false, false);
    }

    v8f c = c0 + c1;
    if (lane == 0 || lane == 16) {
        float* dst = scores + row_base + ((lane == 0) ? 0 : 8);
        #pragma unroll
        for (int i = 0; i < 8; ++i) dst[i] = c[i];
    }
}

// ---------------------------------------------------------------------------
// Stage 2: per-batch WELDON min/max pooling over R=10 segments + sigmoid.
// One 256-thread block per batch; LDS tree reduction per segment.
// ---------------------------------------------------------------------------
__global__ __launch_bounds__(256) void weldon_pool(
    const float* __restrict__ scores, const long long* __restrict__ lengths,
    float* __restrict__ out)
{
    __shared__ float rmax[256];
    __shared__ float rmin[256];
    __shared__ float s_total;

    const int b = blockIdx.x;
    const int tid = threadIdx.x;

    long long len = lengths[b];
    if (len < 1)  len = 1;
    if (len > LL) len = LL;

    const float* srow = scores + (long long)b * LL;

    if (tid == 0) s_total = 0.0f;
    __syncthreads();

    for (int j = 0; j < RR; ++j) {
        const int start = (int)(((long long)j * len) / RR);
        const int end   = (int)(((long long)(j + 1) * len + (RR - 1)) / RR);

        float mx = -3.402823466e38f;   // finfo(float32).min
        float mn =  3.402823466e38f;
        for (int l = start + tid; l < end; l += 256) {
            float v = srow[l];
            mx = fmaxf(mx, v);
            mn = fminf(mn, v);
        }
        rmax[tid] = mx;
        rmin[tid] = mn;
        __syncthreads();
        for (int off = 128; off > 0; off >>= 1) {
            if (tid < off) {
                rmax[tid] = fmaxf(rmax[tid], rmax[tid + off]);
                rmin[tid] = fminf(rmin[tid], rmin[tid + off]);
            }
            __syncthreads();
        }
        if (tid == 0) s_total += rmax[0] + rmin[0];
        __syncthreads();
    }

    if (tid == 0) out[b] = 1.0f / (1.0f + expf(-s_total));
}

extern "C" void kernel_launch(void* const* d_in, const int* in_sizes, int n_in,
                              void* d_out, int out_size, void* d_ws, size_t ws_size,
                              hipStream_t stream) {
    const float*     x       = (const float*)d_in[0];      // (32, 4000, 2048) f32
    const float*     w       = (const float*)d_in[1];      // (2048,) f32
    const long long* lengths = (const long long*)d_in[2];  // (32,) int64
    float*           out     = (float*)d_out;              // (32,) f32
    float*           scores  = (float*)d_ws;               // 32*4000 f32 = 512 KB

    const int tiles  = (BB * LL) / 16;                 // 8000
    const int blocks = tiles / WAVES_PER_BLOCK;        // 1000

    weldon_gemv_wmma<<<blocks, 256, 0, stream>>>(x, w, scores);
    weldon_pool<<<BB, 256, 0, stream>>>(scores, lengths, out);
}